// WDPOLoss_19164144075347
// MI455X (gfx1250) — compile-verified
//
#include <hip/hip_runtime.h>
#include <hip/hip_bf16.h>
#include <stdint.h>

// Problem constants (fixed by the reference).
#define BATCH   32
#define SEQ     2048
#define DIM     1024
#define NPS     (SEQ * DIM)        // elements per sample = 2,097,152
#define CHUNKS  64
#define CHUNK   (NPS / CHUNKS)     // 32,768 elements per block
#define BETA_C  0.1f
#define RHO_C   1.0f

typedef __attribute__((ext_vector_type(2)))  float        v2f;
typedef __attribute__((ext_vector_type(4)))  float        v4f;
typedef __attribute__((ext_vector_type(8)))  float        v8f;
typedef __attribute__((ext_vector_type(4)))  unsigned int v4u;
typedef __attribute__((ext_vector_type(4)))  int          v4i;
typedef __attribute__((ext_vector_type(8)))  int          v8i;

// ---------------------------------------------------------------------------
// Wave32 reduction of 64 per-lane partials (2 per lane) via one f32 WMMA:
// A = 16x4 matrix formed by the wave's v2f accumulators, B = all-ones 4x16.
// D[m][n] = sum_k A[m][k]; summing the 8 D VGPRs gives rows 0..7 on lanes
// 0-15 and rows 8..15 on lanes 16-31; one xor-16 bpermute-add completes it.
// ---------------------------------------------------------------------------
__device__ __forceinline__ float wave_reduce_wmma(v2f acc) {
    v2f ones;
    ones.x = 1.0f;
    ones.y = 1.0f;
    v8f c = {};
    c = __builtin_amdgcn_wmma_f32_16x16x4_f32(
        /*neg_a=*/false, acc, /*neg_b=*/false, ones,
        /*c_mod=*/(short)0, c, /*reuse_a=*/false, /*reuse_b=*/false);
    float s = c[0] + c[1] + c[2] + c[3] + c[4] + c[5] + c[6] + c[7];
    int lane = (int)(threadIdx.x & 31u);
    int o = __builtin_amdgcn_ds_bpermute(((lane ^ 16) << 2), __float_as_int(s));
    return s + __int_as_float(o);
}

__device__ __forceinline__ float xor_reduce32(float x) {
    int lane = (int)(threadIdx.x & 31u);
#pragma unroll
    for (int m = 16; m >= 1; m >>= 1) {
        int o = __builtin_amdgcn_ds_bpermute(((lane ^ m) << 2), __float_as_int(x));
        x += __int_as_float(o);
    }
    return x;
}

// ---------------------------------------------------------------------------
// Pass 1: partial dot products  partial[t][b][chunk] = sum_i e_t[b, i] * w[i%D]
// Grid: (CHUNKS, BATCH, 2), 256 threads (8 waves).
// Each loop iteration the block consumes exactly one contiguous 1024-float
// row, so thread t always multiplies against w[4t .. 4t+3].
// ---------------------------------------------------------------------------
__global__ __launch_bounds__(256) void wdpo_pass1(
    const float* __restrict__ emb1, const float* __restrict__ emb2,
    const float* __restrict__ w, float* __restrict__ partials) {

    __shared__ __align__(16) float smem_w[DIM];
    __shared__ float smem_red[8];

    const unsigned tid = threadIdx.x;

    // --- Stage w (4 KB) into LDS with the Tensor Data Mover (wave 0 only) ---
    if ((tid >> 5) == 0) {
        uint32_t lds_off = (uint32_t)(uintptr_t)(&smem_w[0]);   // generic->LDS offset
        uint64_t ga = (uint64_t)(uintptr_t)w;

        v4u g0;
        g0[0] = 1u;                                    // count=1, user descriptor
        g0[1] = lds_off;                               // lds_addr
        g0[2] = (unsigned)(ga & 0xFFFFFFFFu);          // global_addr[31:0]
        g0[3] = (unsigned)((ga >> 32) & 0x01FFFFFFu)   // global_addr[56:32]
              | 0x80000000u;                           // type = 2 ("image")

        v8i g1;
        g1[0] = (int)(2u << 16);        // data_size = 4 bytes, wg_mask = 0
        g1[1] = (int)(1024u << 16);     // tensor_dim0 = 1024 (low 16 bits)
        g1[2] = (int)(1u << 16);        // tensor_dim1 = 1
        g1[3] = (int)(1024u << 16);     // tile_dim0 = 1024
        g1[4] = 1;                      // tile_dim1 = 1, tile_dim2 = 0
        g1[5] = 1024;                   // tensor_dim0_stride = 1024
        g1[6] = 0;
        g1[7] = 0;

        v4i g2 = {0, 0, 0, 0};
        v4i g3 = {0, 0, 0, 0};
        v8i g4 = {0, 0, 0, 0, 0, 0, 0, 0};

        __builtin_amdgcn_tensor_load_to_lds(g0, g1, g2, g3, g4, /*cpol=*/0);
        __builtin_amdgcn_s_wait_tensorcnt(0);
    }
    __syncthreads();

    const v4f wv = *reinterpret_cast<const v4f*>(&smem_w[tid * 4u]);

    const float* e = (blockIdx.z == 0) ? emb1 : emb2;
    const size_t base = (size_t)blockIdx.y * (size_t)NPS + (size_t)blockIdx.x * (size_t)CHUNK;
    const v4f* __restrict__ p = reinterpret_cast<const v4f*>(e + base);

    v2f acc;
    acc.x = 0.0f;
    acc.y = 0.0f;
#pragma unroll 8
    for (int i = 0; i < CHUNK / DIM; ++i) {            // 32 iterations
        v4f v = __builtin_nontemporal_load(&p[(unsigned)i * 256u + tid]);
        acc.x = fmaf(v.x, wv.x, fmaf(v.z, wv.z, acc.x));
        acc.y = fmaf(v.y, wv.y, fmaf(v.w, wv.w, acc.y));
    }

    // Wave-level reduction via WMMA, then cross-wave via LDS.
    float wsum = wave_reduce_wmma(acc);
    if ((tid & 31u) == 0u) smem_red[tid >> 5] = wsum;
    __syncthreads();
    if (tid == 0) {
        float t = 0.0f;
#pragma unroll
        for (int i = 0; i < 8; ++i) t += smem_red[i];
        partials[((size_t)blockIdx.z * BATCH + blockIdx.y) * CHUNKS + blockIdx.x] = t;
    }
}

// ---------------------------------------------------------------------------
// Pass 2: one wave. Lane b folds the 64 chunk partials of sample b for both
// tensors, evaluates the closed-form WDPO loss and gradient-norm term, then
// cross-lane reduces and writes the final scalar.
// ---------------------------------------------------------------------------
__device__ __forceinline__ float softplus_f(float x) {
    return fmaxf(x, 0.0f) + log1pf(expf(-fabsf(x)));
}

__global__ __launch_bounds__(32) void wdpo_pass2(
    const float* __restrict__ partials, const float* __restrict__ w,
    const float* __restrict__ ref1, const float* __restrict__ ref2,
    const float* __restrict__ pref, float* __restrict__ out) {

    const int lane = (int)(threadIdx.x & 31u);

    float lp1 = 0.0f, lp2 = 0.0f;
#pragma unroll 8
    for (int c = 0; c < CHUNKS; ++c) {
        lp1 += partials[(size_t)lane * CHUNKS + c];
        lp2 += partials[(size_t)(BATCH + lane) * CHUNKS + c];
    }

    // ||w||^2 : lane l sums w[l + 32k]^2 (coalesced), reduced below.
    float wsq = 0.0f;
#pragma unroll 8
    for (int k = 0; k < DIM / 32; ++k) {
        float ww = w[lane + 32 * k];
        wsq = fmaf(ww, ww, wsq);
    }

    const float p  = pref[lane];
    const float h  = (lp1 - ref1[lane]) - (lp2 - ref2[lane]);
    const float z  = BETA_C * h;
    const float l1 = softplus_f(-z);                 // -log_sigmoid(+z)
    const float l2 = softplus_f(z);                  // -log_sigmoid(-z)
    const float ind = p * l1 + (1.0f - p) * l2;
    const float sig = 1.0f / (1.0f + expf(-z));
    const float cb  = sig - p;
    const float gterm = cb * cb;

    const float sum_ind = xor_reduce32(ind);
    const float sum_gt  = xor_reduce32(gterm);
    const float wnorm2  = xor_reduce32(wsq);

    if (lane == 0) {
        // grad_norm_sq[b] = 2*beta^2*(sig-p)^2 * S * ||w||^2
        const float gns_mean = 2.0f * BETA_C * BETA_C * (float)SEQ * wnorm2 * (sum_gt / (float)BATCH);
        out[0] = sum_ind / (float)BATCH + RHO_C * sqrtf(gns_mean);
    }
}

// ---------------------------------------------------------------------------
extern "C" void kernel_launch(void* const* d_in, const int* in_sizes, int n_in,
                              void* d_out, int out_size, void* d_ws, size_t ws_size,
                              hipStream_t stream) {
    const float* emb1 = (const float*)d_in[0];
    const float* emb2 = (const float*)d_in[1];
    const float* w    = (const float*)d_in[2];
    const float* r1   = (const float*)d_in[3];
    const float* r2   = (const float*)d_in[4];
    const float* pr   = (const float*)d_in[5];
    float* out        = (float*)d_out;
    float* partials   = (float*)d_ws;   // 2 * BATCH * CHUNKS floats = 16 KB

    dim3 grid(CHUNKS, BATCH, 2);
    wdpo_pass1<<<grid, 256, 0, stream>>>(emb1, emb2, w, partials);
    wdpo_pass2<<<1, 32, 0, stream>>>(partials, w, r1, r2, pr, out);
}